// LGCNEncoder_90881507983942
// MI455X (gfx1250) — compile-verified
//
#include <hip/hip_runtime.h>
#include <hip/hip_bf16.h>
#include <stdint.h>

#define WAVE            32
#define CHUNK           32      // edges staged per wave per step (one per lane)
#define WAVES_PER_BLOCK 8
#define EMB             64

typedef __attribute__((ext_vector_type(2))) float v2f;
typedef __attribute__((ext_vector_type(8))) float v8f;

// Low 32 bits of a flat shared-space address == LDS byte offset (aperture layout).
__device__ __forceinline__ unsigned lds_off_of(const void* p) {
    return (unsigned)(unsigned long long)(uintptr_t)p;
}

// CDNA5 async copy global->LDS (ASYNCcnt path). VDST = LDS byte offset VGPR,
// VADDR = 64-bit global address VGPR pair, no SADDR.
__device__ __forceinline__ void async_load_dw_to_lds(unsigned lds_off,
                                                     unsigned long long gaddr) {
    asm volatile("global_load_async_to_lds_b32 %0, %1, off"
                 :: "v"(lds_off), "v"(gaddr)
                 : "memory");
}

__device__ __forceinline__ void wait_async0() {
    asm volatile("s_wait_asynccnt 0x0" ::: "memory");
}

// ---------------------------------------------------------------------------
// COO SpMM scatter: one wave per edge; lane l handles elements 2l, 2l+1.
// Edge row/col/val are streamed into per-wave LDS slots via async-to-LDS,
// then consumed as uniform broadcast DS reads. Gathers are coalesced 256B
// rows; scatters are no-return global_atomic_add_f32 (table is L2-resident).
// ---------------------------------------------------------------------------
__global__ __launch_bounds__(256) void lgcn_spmm_scatter(
    const int* __restrict__ erow, const int* __restrict__ ecol,
    const float* __restrict__ eval, const float* __restrict__ src,
    float* __restrict__ dst, int ne)
{
    __shared__ int   sRow[WAVES_PER_BLOCK][CHUNK];
    __shared__ int   sCol[WAVES_PER_BLOCK][CHUNK];
    __shared__ float sVal[WAVES_PER_BLOCK][CHUNK];

    const int lane = threadIdx.x & (WAVE - 1);
    const int wv   = threadIdx.x >> 5;
    const int gw   = (int)((blockIdx.x * blockDim.x + threadIdx.x) >> 5);
    const int nw   = (int)((gridDim.x * blockDim.x) >> 5);
    const int nchunks = (ne + CHUNK - 1) / CHUNK;

    for (int c = gw; c < nchunks; c += nw) {
        const int base = c * CHUNK;
        const int cnt  = (ne - base < CHUNK) ? (ne - base) : CHUNK;

        if (lane < cnt) {  // EXEC-masked async issues; per-wave ASYNCcnt
            async_load_dw_to_lds(lds_off_of(&sRow[wv][lane]),
                                 (unsigned long long)(uintptr_t)(erow + base + lane));
            async_load_dw_to_lds(lds_off_of(&sCol[wv][lane]),
                                 (unsigned long long)(uintptr_t)(ecol + base + lane));
            async_load_dw_to_lds(lds_off_of(&sVal[wv][lane]),
                                 (unsigned long long)(uintptr_t)(eval + base + lane));
        }
        wait_async0();  // wave-private wait; no barrier needed

        for (int j = 0; j < cnt; ++j) {
            const int   r  = sRow[wv][j];   // uniform broadcast DS reads
            const int   ci = sCol[wv][j];
            const float v  = sVal[wv][j];

            const float2 x = *(const float2*)(src + (size_t)ci * EMB + lane * 2);
            float* o = dst + (size_t)r * EMB + lane * 2;
            atomicAdd(o,     v * x.x);   // global_atomic_add_f32, no return
            atomicAdd(o + 1, v * x.y);
        }
    }
}

// ---------------------------------------------------------------------------
// Mean over 4 layer snapshots via V_WMMA_F32_16X16X4_F32:
//   D = A(16x4 of 0.25) x B(4x16 data) ;  D[0, n] = 0.25 * sum_k stacked[k, j0+n]
// B layout (4x16, 2 VGPRs, wave32): lanes 0-15 carry rows K=0 (b.x), K=1 (b.y);
// lanes 16-31 carry rows K=2 (b.x), K=3 (b.y). A is constant so its layout is
// immaterial. D VGPR0, lanes 0-15 = (M=0, N=lane) -> 16 means per instruction.
// ---------------------------------------------------------------------------
__global__ __launch_bounds__(256) void lgcn_mean4(
    const float* __restrict__ stacked, float* __restrict__ outf, long nv)
{
    const int  lane = threadIdx.x & (WAVE - 1);
    const long gw   = (long)((blockIdx.x * blockDim.x + threadIdx.x) >> 5);
    const long nw   = (long)((gridDim.x * (long)blockDim.x) >> 5);
    const int  l0   = (lane < 16) ? 0 : 2;
    const int  k    = lane & 15;

#if __has_builtin(__builtin_amdgcn_wmma_f32_16x16x4_f32)
    for (long j0 = gw * 16; j0 < nv; j0 += nw * 16) {   // uniform loop, EXEC all-1s
        v2f b;
        b.x = stacked[(long)l0 * nv + j0 + k];
        b.y = stacked[(long)(l0 + 1) * nv + j0 + k];
        v2f a; a.x = 0.25f; a.y = 0.25f;
        v8f c = {};
        c = __builtin_amdgcn_wmma_f32_16x16x4_f32(
                /*neg_a=*/false, a, /*neg_b=*/false, b,
                /*c_mod=*/(short)0, c, /*reuse_a=*/false, /*reuse_b=*/false);
        if (lane < 16) outf[j0 + k] = c[0];
    }
#else
    // VALU fallback (keeps the file compiling if the builtin is absent)
    long i = (long)blockIdx.x * blockDim.x + threadIdx.x;
    long s = (long)gridDim.x * blockDim.x;
    for (; i < nv; i += s)
        outf[i] = 0.25f * (stacked[i] + stacked[nv + i] +
                           stacked[2 * nv + i] + stacked[3 * nv + i]);
#endif
}

__global__ void lgcn_zero(float4* __restrict__ p, long n4) {
    long i = (long)blockIdx.x * blockDim.x + threadIdx.x;
    long s = (long)gridDim.x * blockDim.x;
    float4 z; z.x = 0.f; z.y = 0.f; z.z = 0.f; z.w = 0.f;
    for (; i < n4; i += s) p[i] = z;
}

extern "C" void kernel_launch(void* const* d_in, const int* in_sizes, int n_in,
                              void* d_out, int out_size, void* d_ws, size_t ws_size,
                              hipStream_t stream) {
    const float* user_emb = (const float*)d_in[0];
    const float* item_emb = (const float*)d_in[1];
    const int*   erow     = (const int*)d_in[2];
    const int*   ecol     = (const int*)d_in[3];
    const float* evalp    = (const float*)d_in[4];

    const long u64 = in_sizes[0];          // USER_NUM * 64
    const long i64 = in_sizes[1];          // ITEM_NUM * 64
    const int  ne  = in_sizes[2];          // 2 * N_INTERACTIONS
    const long nv  = u64 + i64;            // N * 64 (multiple of 64)

    float* outp   = (float*)d_out;
    float* finalp = outp;                  // [nv]  (users ‖ items, contiguous)
    float* st     = outp + nv;             // [4][nv] stacked snapshots

    // stacked[0] = concat(user_emb, item_emb)  (in-place in d_out)
    hipMemcpyAsync(st,       user_emb, (size_t)u64 * sizeof(float),
                   hipMemcpyDeviceToDevice, stream);
    hipMemcpyAsync(st + u64, item_emb, (size_t)i64 * sizeof(float),
                   hipMemcpyDeviceToDevice, stream);

    const int nchunks = (ne + CHUNK - 1) / CHUNK;
    int spmmBlocks = (nchunks + WAVES_PER_BLOCK - 1) / WAVES_PER_BLOCK;
    if (spmmBlocks > 4096) spmmBlocks = 4096;

    const long n4 = nv / 4;
    int zBlocks = (int)((n4 + 255) / 256);
    if (zBlocks > 8192) zBlocks = 8192;

    for (int l = 1; l <= 3; ++l) {
        float*       dstL = st + (long)l * nv;
        const float* srcL = st + (long)(l - 1) * nv;
        lgcn_zero<<<zBlocks, 256, 0, stream>>>((float4*)dstL, n4);
        lgcn_spmm_scatter<<<spmmBlocks, 256, 0, stream>>>(erow, ecol, evalp,
                                                          srcL, dstL, ne);
    }

    lgcn_mean4<<<2048, 256, 0, stream>>>(st, finalp, nv);
}